// _FrequencyBandAverager_29686813949953
// MI455X (gfx1250) — compile-verified
//
#include <hip/hip_runtime.h>

// CDNA5 / gfx1250 — wave32, V_WMMA_F32_16X16X4_F32.
// out[bt,g,col] = sum_f mask[g,f]*x[bt,f,col] / count[g]
// One wave = one (b,t) slice x one 16-wide column tile. K=129 = 32 WMMA K4
// steps (f=0..127) + scalar epilogue for f=128.
typedef __attribute__((ext_vector_type(2))) float v2f;
typedef __attribute__((ext_vector_type(8))) float v8f;

#define NFREQ  129
#define NBAND  8
#define CH     1024        // 32*32 flattened channels
#define NTILE  64          // CH/16
#define NBT    400         // 8*50
#define WPB    8           // waves per 256-thread block
#define KSTEPS 32          // f = 0..127 via WMMA

__global__ __launch_bounds__(256)
void _FrequencyBandAverager_wmma_kernel(const float* __restrict__ x,
                                        const float* __restrict__ mask,
                                        float* __restrict__ out)
{
    // A fragments, per WMMA step, in exact lane layout: sA[ks][lane][0..1]
    __shared__ __align__(16) float sA[KSTEPS * 64];   // 8 KB
    __shared__ float sRc[NBAND];                      // 1/count[g]
    __shared__ float sTail[NBAND];                    // mask[g][128]/count[g]

    const int tid = threadIdx.x;

    // --- Block prologue: band counts + f=128 tail coefficients -------------
    if (tid < NBAND) {
        const float* mr = mask + tid * NFREQ;
        float cnt = 0.f;
        for (int f = 0; f < NFREQ; ++f) cnt += mr[f];
        const float rc = 1.0f / cnt;
        sRc[tid]   = rc;
        sTail[tid] = mr[128] * rc;
    }
    __syncthreads();

    // --- Stage scaled A fragments (rows 8..15 zero-padded) -----------------
    // Entry (ks, lane l, v): M = l%16, K = ks*4 + 2*(l/16) + v  (K <= 127)
    for (int idx = tid; idx < KSTEPS * 64; idx += 256) {
        const int ks = idx >> 6;
        const int lv = idx & 63;
        const int l  = lv >> 1;
        const int v  = lv & 1;
        const int m  = l & 15;
        const int k  = ks * 4 + ((l >> 4) << 1) + v;
        sA[idx] = (m < NBAND) ? mask[m * NFREQ + k] * sRc[m] : 0.0f;
    }
    __syncthreads();

    // --- Per-wave GEMM tile -------------------------------------------------
    const int lane  = tid & 31;
    const int w     = blockIdx.x * WPB + (tid >> 5);
    const int bt    = w >> 6;
    const int nt    = w & 63;
    const int m     = lane & 15;     // A row / B column
    const int khalf = lane >> 4;     // K sub-pair select

    // B element base: column (nt*16 + m), rows k0 = ks*4 + 2*khalf (+1)
    const float* xw = x + (size_t)bt * NFREQ * CH + (size_t)(khalf * 2) * CH
                        + nt * 16 + m;

    v8f c = {};
    #pragma unroll
    for (int ks = 0; ks < KSTEPS; ++ks) {
        const v2f a = *(const v2f*)(sA + ks * 64 + lane * 2);  // ds_load_b64
        v2f b;
        b.x = xw[(size_t)ks * 4 * CH];        // row k0,   64B/half-wave
        b.y = xw[(size_t)ks * 4 * CH + CH];   // row k0+1
        c = __builtin_amdgcn_wmma_f32_16x16x4_f32(
                /*neg_a=*/false, a, /*neg_b=*/false, b,
                /*c_mod=*/(short)0, c, /*reuse_a=*/false, /*reuse_b=*/false);
    }

    // --- Epilogue: f = 128 contribution + store (lanes 0-15 hold M=0..7) ---
    if (lane < 16) {
        const float xv = x[(size_t)bt * NFREQ * CH + (size_t)128 * CH
                           + nt * 16 + lane];
        float* o = out + (size_t)bt * NBAND * CH + nt * 16 + lane;
        #pragma unroll
        for (int g = 0; g < NBAND; ++g)
            o[(size_t)g * CH] = c[g] + sTail[g] * xv;
    }
}

extern "C" void kernel_launch(void* const* d_in, const int* in_sizes, int n_in,
                              void* d_out, int out_size, void* d_ws, size_t ws_size,
                              hipStream_t stream)
{
    const float* x    = (const float*)d_in[0];   // (8,50,129,32,32) f32
    const float* mask = (const float*)d_in[1];   // (8,129) f32
    float* out        = (float*)d_out;           // (8,50,8,32,32) f32

    const int grid = (NBT * NTILE) / WPB;        // 3200 blocks of 256
    _FrequencyBandAverager_wmma_kernel<<<grid, 256, 0, stream>>>(x, mask, out);
}